// Attention_50087908606296
// MI455X (gfx1250) — compile-verified
//
#include <hip/hip_runtime.h>

typedef __attribute__((ext_vector_type(2))) float v2f;
typedef __attribute__((ext_vector_type(4))) float v4f;
typedef __attribute__((ext_vector_type(8))) float v8f;

constexpr int kB       = 32;    // batch
constexpr int kS       = 4096;  // sequence
constexpr int kH       = 1024;  // hidden
constexpr int kChunk   = 256;   // S-rows per workgroup
constexpr int kNChunk  = kS / kChunk;   // 16
constexpr int kThreads = 256;           // 8 wave32s

// ---------------------------------------------------------------------------
// Pass 1: one workgroup per (batch, S-chunk).
//   Phase A: scores via V_WMMA_F32_16X16X4_F32 (enc rows x dec), dec in LDS.
//   Phase B: chunk-local softmax stats (m, Z) + e_s = exp(score - m).
//   Phase C: partial context = sum_s e_s * enc[s, :]  (chunk re-read, L2-hot).
// ---------------------------------------------------------------------------
__global__ __launch_bounds__(kThreads)
void attn_pass1(const float* __restrict__ hidden,
                const float* __restrict__ enc,
                float* __restrict__ out,      // [kB*kH ctx][kB*kS weights]
                float* __restrict__ ws_ctx,   // kB*kNChunk*kH partial contexts
                float* __restrict__ ws_mz)    // kB*kNChunk*2 (m, Z)
{
    const int chunk = blockIdx.x;
    const int b     = blockIdx.y;
    const int tid   = threadIdx.x;
    const int s0    = chunk * kChunk;

    const float* dec  = hidden + (size_t)kB * kH + (size_t)b * kH;  // hidden[-1][b]
    const float* encb = enc + (size_t)b * kS * kH;

    __shared__ float lds_dec[kH];     // 4 KB
    __shared__ float lds_sc[kChunk];  // scores, then e_s
    __shared__ float red[kThreads];   // reduction scratch

    // stage dec[b] into LDS (256 threads x float4)
    *(v4f*)&lds_dec[tid * 4] = *(const v4f*)(dec + tid * 4);
    __syncthreads();

    // ---------------- Phase A: WMMA scores ----------------
    const int wave = tid >> 5;        // 0..7
    const int lane = tid & 31;
    const int half = lane >> 4;       // K-half selector (ISA A/B layout)
    const int mrow = lane & 15;       // matrix row within tile

    for (int t = wave; t < kChunk / 16; t += kThreads / 32) {   // 2 tiles/wave
        const float* arow = encb + (size_t)(s0 + t * 16 + mrow) * kH;
        v8f c = {};
        #pragma unroll 8
        for (int k = 0; k < kH; k += 4) {
            const int kk = k + 2 * half;
            v2f a = *(const v2f*)(arow + kk);           // A[m, kk..kk+1]
            v2f bcol;                                   // dec replicated over N
            bcol.x = lds_dec[kk];
            bcol.y = lds_dec[kk + 1];
            c = __builtin_amdgcn_wmma_f32_16x16x4_f32(
                    /*neg_a=*/false, a, /*neg_b=*/false, bcol,
                    /*c_mod=*/(short)0, c, /*reuse_a=*/false, /*reuse_b=*/false);
        }
        // D layout: VGPR v holds row (v + 8*half) at col (lane&15); take col 0.
        if ((lane & 15) == 0) {
            #pragma unroll
            for (int v = 0; v < 8; ++v)
                lds_sc[t * 16 + v + 8 * half] = c[v];
        }
    }
    __syncthreads();

    // ---------------- Phase B: local softmax stats ----------------
    const float sval = lds_sc[tid];
    red[tid] = sval;
    __syncthreads();
    for (int off = kThreads / 2; off > 0; off >>= 1) {
        if (tid < off) red[tid] = fmaxf(red[tid], red[tid + off]);
        __syncthreads();
    }
    const float m = red[0];
    __syncthreads();

    const float e = __expf(sval - m);
    lds_sc[tid] = e;           // e_s for Phase C (own slot only)
    red[tid]    = e;
    __syncthreads();
    for (int off = kThreads / 2; off > 0; off >>= 1) {
        if (tid < off) red[tid] += red[tid + off];
        __syncthreads();
    }
    const float Z = red[0];

    // stash un-normalized e into the weights slot of d_out (pass 2 rescales)
    out[(size_t)kB * kH + (size_t)b * kS + s0 + tid] = e;
    if (tid == 0) {
        ws_mz[((size_t)b * kNChunk + chunk) * 2 + 0] = m;
        ws_mz[((size_t)b * kNChunk + chunk) * 2 + 1] = Z;
    }

    // ---------------- Phase C: partial context ----------------
    const int h0 = tid * 4;
    const float* p = encb + (size_t)s0 * kH + h0;
    float ax = 0.f, ay = 0.f, az = 0.f, aw = 0.f;
    #pragma unroll 4
    for (int s = 0; s < kChunk; ++s) {
        v4f ev = *(const v4f*)(p + (size_t)s * kH);   // L2-hot re-read
        const float w = lds_sc[s];
        ax += w * ev.x; ay += w * ev.y; az += w * ev.z; aw += w * ev.w;
    }
    v4f acc; acc.x = ax; acc.y = ay; acc.z = az; acc.w = aw;
    *(v4f*)(ws_ctx + ((size_t)b * kNChunk + chunk) * kH + h0) = acc;
}

// ---------------------------------------------------------------------------
// Pass 2: one block per batch. Merge chunk stats (log-sum-exp rescale),
// write context, normalize weights in place.
// ---------------------------------------------------------------------------
__global__ __launch_bounds__(kThreads)
void attn_pass2(float* __restrict__ out,
                const float* __restrict__ ws_ctx,
                const float* __restrict__ ws_mz)
{
    const int b   = blockIdx.x;
    const int tid = threadIdx.x;

    __shared__ float mz[kNChunk * 2];
    __shared__ float scale[kNChunk];

    if (tid < kNChunk * 2) mz[tid] = ws_mz[(size_t)b * kNChunk * 2 + tid];
    __syncthreads();

    if (tid == 0) {
        float m = -3.402823466e38f;
        for (int c = 0; c < kNChunk; ++c) m = fmaxf(m, mz[2 * c]);
        float Z = 0.f;
        for (int c = 0; c < kNChunk; ++c) Z += mz[2 * c + 1] * __expf(mz[2 * c] - m);
        const float invZ = 1.0f / Z;
        for (int c = 0; c < kNChunk; ++c) scale[c] = __expf(mz[2 * c] - m) * invZ;
    }
    __syncthreads();

    // context[b, h] = sum_c scale[c] * ctx_partial[c, h]
    const int h0 = tid * 4;
    float ax = 0.f, ay = 0.f, az = 0.f, aw = 0.f;
    #pragma unroll
    for (int c = 0; c < kNChunk; ++c) {
        v4f v = *(const v4f*)(ws_ctx + ((size_t)b * kNChunk + c) * kH + h0);
        const float sc = scale[c];
        ax += sc * v.x; ay += sc * v.y; az += sc * v.z; aw += sc * v.w;
    }
    v4f acc; acc.x = ax; acc.y = ay; acc.z = az; acc.w = aw;
    *(v4f*)(out + (size_t)b * kH + h0) = acc;

    // weights[b, s] = e_s * scale[chunk(s)]
    float* w = out + (size_t)kB * kH + (size_t)b * kS;
    #pragma unroll
    for (int i = 0; i < kS / kThreads; ++i) {
        const int s = tid + i * kThreads;
        w[s] = w[s] * scale[s / kChunk];
    }
}

extern "C" void kernel_launch(void* const* d_in, const int* in_sizes, int n_in,
                              void* d_out, int out_size, void* d_ws, size_t ws_size,
                              hipStream_t stream) {
    const float* hidden = (const float*)d_in[0];  // (2, 32, 1024) f32
    const float* enc    = (const float*)d_in[1];  // (32, 4096, 1024) f32
    float* out = (float*)d_out;                   // [32*1024 ctx][32*4096 weights]

    float* ws_ctx = (float*)d_ws;                              // 2 MiB
    float* ws_mz  = ws_ctx + (size_t)kB * kNChunk * kH;        // 4 KiB

    dim3 grid1(kNChunk, kB), blk(kThreads);
    attn_pass1<<<grid1, blk, 0, stream>>>(hidden, enc, out, ws_ctx, ws_mz);
    attn_pass2<<<dim3(kB), blk, 0, stream>>>(out, ws_ctx, ws_mz);
}